// GraphEncoder_84413287236170
// MI455X (gfx1250) — compile-verified
//
#include <hip/hip_runtime.h>
#include <hip/hip_bf16.h>

#define NN   2048
#define BB   32
#define KNN_ 16
// degree = KNN + self-loop = 17 -> gcn_norm constant 1/17

typedef __attribute__((ext_vector_type(2))) float v2f;
typedef __attribute__((ext_vector_type(8))) float v8f;

// ---------------------------------------------------------------------------
// Kernel 1: brute-force KNN (k=16, self excluded).
// Block = 256 threads handles 256 query nodes of one sample; all 2048 points
// of the sample staged in LDS (SoA, 24KB). Inner loop is an LDS broadcast.
// Top-16 kept sorted in registers; branchless unrolled insertion.
// ---------------------------------------------------------------------------
__global__ __launch_bounds__(256) void knn_kernel(const float* __restrict__ coords,
                                                  int* __restrict__ nbr) {
    const int b   = blockIdx.y;
    const int tid = threadIdx.x;
    const int q   = blockIdx.x * 256 + tid;

    __shared__ float sx[NN], sy[NN], sz[NN];
    const float* cb = coords + (size_t)b * NN * 3;
    for (int i = tid; i < NN; i += 256) {
        sx[i] = cb[i * 3 + 0];
        sy[i] = cb[i * 3 + 1];
        sz[i] = cb[i * 3 + 2];
    }
    __syncthreads();

    const float qx = sx[q], qy = sy[q], qz = sz[q];

    float dist[KNN_];
    int   idx[KNN_];
#pragma unroll
    for (int t = 0; t < KNN_; ++t) { dist[t] = 3.0e38f; idx[t] = q; }

    for (int j = 0; j < NN; ++j) {
        float dx = sx[j] - qx;
        float dy = sy[j] - qy;
        float dz = sz[j] - qz;
        float d  = fmaf(dx, dx, fmaf(dy, dy, dz * dz));
        d = (j == q) ? 3.4e38f : d;          // exclude self (loop=False)
        if (d < dist[KNN_ - 1]) {
            // sorted (ascending) insertion, fully unrolled, select-only.
            // slot t reads OLD dist[t-1]/dist[t]; descending t keeps them live.
#pragma unroll
            for (int t = KNN_ - 1; t >= 1; --t) {
                bool c1 = d < dist[t - 1];
                bool c2 = d < dist[t];
                float nd = c1 ? dist[t - 1] : (c2 ? d : dist[t]);
                int   ni = c1 ? idx[t - 1]  : (c2 ? j : idx[t]);
                dist[t] = nd;
                idx[t]  = ni;
            }
            bool c0 = d < dist[0];
            idx[0]  = c0 ? j : idx[0];
            dist[0] = c0 ? d : dist[0];
        }
    }

    int* nb = nbr + ((size_t)b * NN + q) * KNN_;
#pragma unroll
    for (int t = 0; t < KNN_; ++t) nb[t] = idx[t];
}

// ---------------------------------------------------------------------------
// Kernel 2: layer 1. aggregate coords over nbr+self (/17), then 3->64 matmul
// + bias + relu. K=3 so plain FMA; W1/b1 in LDS; coalesced x1 writes.
// ---------------------------------------------------------------------------
__global__ __launch_bounds__(256) void layer1_kernel(const float* __restrict__ coords,
                                                     const int* __restrict__ nbr,
                                                     const float* __restrict__ W1,
                                                     const float* __restrict__ b1,
                                                     float* __restrict__ x1) {
    const int b     = blockIdx.y;
    const int tid   = threadIdx.x;
    const int node0 = blockIdx.x * 256;
    const int q     = node0 + tid;

    __shared__ float sW1[3 * 64];
    __shared__ float sb1[64];
    __shared__ float cag[256][3];

    if (tid < 192) sW1[tid] = W1[tid];
    if (tid < 64)  sb1[tid] = b1[tid];

    const float* cb = coords + (size_t)b * NN * 3;
    const int*   nb = nbr + ((size_t)b * NN + q) * KNN_;

    float ax = cb[q * 3 + 0];
    float ay = cb[q * 3 + 1];
    float az = cb[q * 3 + 2];
#pragma unroll 4
    for (int u = 0; u < KNN_; ++u) {
        int j = nb[u];
        ax += cb[j * 3 + 0];
        ay += cb[j * 3 + 1];
        az += cb[j * 3 + 2];
    }
    const float inv = 1.0f / 17.0f;
    cag[tid][0] = ax * inv;
    cag[tid][1] = ay * inv;
    cag[tid][2] = az * inv;
    __syncthreads();

    // 256 threads -> (4 nodes x 64 channels) per pass, 64 passes, coalesced.
    float* xb  = x1 + ((size_t)b * NN + node0) * 64;
    const int ch   = tid & 63;
    const int nsub = tid >> 6;                  // 0..3
    const float w0 = sW1[0 * 64 + ch];
    const float w1 = sW1[1 * 64 + ch];
    const float w2 = sW1[2 * 64 + ch];
    const float bc = sb1[ch];
    for (int p = 0; p < 64; ++p) {
        int nloc = p * 4 + nsub;
        float v = fmaf(cag[nloc][0], w0,
                  fmaf(cag[nloc][1], w1,
                  fmaf(cag[nloc][2], w2, bc)));
        xb[(size_t)nloc * 64 + ch] = fmaxf(v, 0.0f);
    }
}

// ---------------------------------------------------------------------------
// Kernel 3: fused layer2 + final projection with fp32 WMMA (16x16x4).
// Block = 256 threads (8 waves) owns 16 nodes of one sample:
//   A: gather-sum x1 over nbr+self -> A1[16][64] in LDS (/17)
//   B: x2 = relu(A1 @ W2 + b2)  -> X2[16][128] in LDS   (8 col-tiles, 1/wave)
//   C: out = X2 @ Wf + bf       -> global               (8 col-tiles, 1/wave)
// Fragment layouts per CDNA5 ISA 7.12.2 (32-bit A 16x4, 32-bit C/D 16x16).
// ---------------------------------------------------------------------------
__global__ __launch_bounds__(256) void gcn2_kernel(const float* __restrict__ x1,
                                                   const int* __restrict__ nbr,
                                                   const float* __restrict__ W2,
                                                   const float* __restrict__ b2,
                                                   const float* __restrict__ Wf,
                                                   const float* __restrict__ bf,
                                                   float* __restrict__ out) {
    const int b     = blockIdx.y;
    const int node0 = blockIdx.x * 16;
    const int tid   = threadIdx.x;

    __shared__ int   sidx[16][16];
    __shared__ float A1[16][68];     // 64 + 4 pad: stride%64==4 -> conflict-free
    __shared__ float X2[16][132];    // 128 + 4 pad

    // stage neighbor indices: one per thread
    {
        int nloc = tid >> 4, u = tid & 15;
        sidx[nloc][u] = nbr[((size_t)b * NN + node0 + nloc) * KNN_ + u];
    }
    __syncthreads();

    // aggregate x1 tile: thread = (node, 4-channel group)
    {
        int nloc = tid >> 4;
        int cg   = (tid & 15) * 4;
        const float* xb   = x1 + (size_t)b * NN * 64;
        const float* self = xb + (size_t)(node0 + nloc) * 64 + cg;
        float a0 = self[0], a1 = self[1], a2 = self[2], a3 = self[3];
#pragma unroll 4
        for (int u = 0; u < 16; ++u) {
            const float* p = xb + (size_t)sidx[nloc][u] * 64 + cg;
            a0 += p[0]; a1 += p[1]; a2 += p[2]; a3 += p[3];
        }
        const float inv = 1.0f / 17.0f;
        A1[nloc][cg + 0] = a0 * inv;
        A1[nloc][cg + 1] = a1 * inv;
        A1[nloc][cg + 2] = a2 * inv;
        A1[nloc][cg + 3] = a3 * inv;
    }
    __syncthreads();

    const int wave  = tid >> 5;
    const int lane  = tid & 31;
    const int row   = lane & 15;            // A row / B,C,D column within tile
    const int khalf = (lane >> 4) * 2;      // lanes 0-15 -> K 0,1; 16-31 -> K 2,3
    const int mhi   = (lane >> 4) * 8;      // C/D: M = r (+8 for upper lanes)

    // ---- GEMM1: x2 = relu(A1 @ W2 + b2), W2 row-major [64][128], K=64 ----
    {
        const int n0 = wave * 16;
        v8f acc = {};
#pragma unroll
        for (int kk = 0; kk < 16; ++kk) {
            int k = kk * 4 + khalf;
            v2f a;   a.x   = A1[row][k];
                     a.y   = A1[row][k + 1];
            v2f bf2; bf2.x = W2[(size_t)k * 128 + n0 + row];
                     bf2.y = W2[(size_t)(k + 1) * 128 + n0 + row];
            acc = __builtin_amdgcn_wmma_f32_16x16x4_f32(
                false, a, false, bf2, (short)0, acc, false, false);
        }
#pragma unroll
        for (int r = 0; r < 8; ++r) {
            int m = mhi + r;
            int n = n0 + row;
            X2[m][n] = fmaxf(acc[r] + b2[n], 0.0f);
        }
    }
    __syncthreads();

    // ---- GEMM2: out = X2 @ Wf + bf, Wf row-major [128][128], K=128 ----
    {
        const int n0 = wave * 16;
        v8f acc = {};
#pragma unroll
        for (int kk = 0; kk < 32; ++kk) {
            int k = kk * 4 + khalf;
            v2f a;   a.x   = X2[row][k];
                     a.y   = X2[row][k + 1];
            v2f bf2; bf2.x = Wf[(size_t)k * 128 + n0 + row];
                     bf2.y = Wf[(size_t)(k + 1) * 128 + n0 + row];
            acc = __builtin_amdgcn_wmma_f32_16x16x4_f32(
                false, a, false, bf2, (short)0, acc, false, false);
        }
        float* ob = out + ((size_t)b * NN + node0) * 128;
#pragma unroll
        for (int r = 0; r < 8; ++r) {
            int m = mhi + r;
            int n = n0 + row;
            ob[(size_t)m * 128 + n] = acc[r] + bf[n];
        }
    }
}

// ---------------------------------------------------------------------------
// Launch
// ---------------------------------------------------------------------------
extern "C" void kernel_launch(void* const* d_in, const int* in_sizes, int n_in,
                              void* d_out, int out_size, void* d_ws, size_t ws_size,
                              hipStream_t stream) {
    const float* coords = (const float*)d_in[0];   // [32,2048,3]
    const float* W1     = (const float*)d_in[1];   // [3,64]
    const float* b1     = (const float*)d_in[2];   // [64]
    const float* W2     = (const float*)d_in[3];   // [64,128]
    const float* b2     = (const float*)d_in[4];   // [128]
    const float* Wf     = (const float*)d_in[5];   // [128,128]
    const float* bf     = (const float*)d_in[6];   // [128]
    float* out = (float*)d_out;                    // [32,2048,128]

    char* ws = (char*)d_ws;
    int*   nbr = (int*)ws;                                         // 4 MB
    float* x1  = (float*)(ws + (size_t)BB * NN * KNN_ * sizeof(int)); // 16 MB

    dim3 blk(256);
    knn_kernel   <<<dim3(NN / 256, BB), blk, 0, stream>>>(coords, nbr);
    layer1_kernel<<<dim3(NN / 256, BB), blk, 0, stream>>>(coords, nbr, W1, b1, x1);
    gcn2_kernel  <<<dim3(NN / 16,  BB), blk, 0, stream>>>(x1, nbr, W2, b2, Wf, bf, out);
}